// Model_cosmetics_28570122453214
// MI455X (gfx1250) — compile-verified
//
#include <hip/hip_runtime.h>
#include <float.h>
#include <math.h>

// ---------------- problem constants ----------------
#define NUM_ITEMS 40000
#define NITEMS1   39999      // all_item rows (embedding_weight[1:])
#define D_        300
#define B_        64
#define L_        20
#define S_        5
#define COARSE_   100
#define EXP_      20
#define ROWS_PER_SRC (B_*L_*S_)          // 6400
#define TILES_PER_SRC (ROWS_PER_SRC/16)  // 400
#define EXP_ROWS (B_*L_*COARSE_)         // 128000 rows per source
#define M_TILES_EXP (EXP_ROWS/16)        // 8000
#define N_TILES_EXP 19                   // ceil(300/16)

// d_out layout (fp32 flat, reference return order)
#define OFF_LOSS   0
#define OFF_SCORES 1
#define OFF_SUB    ((size_t)1 + (size_t)B_*NITEMS1)                 // 2559937
#define EXP_STRIDE ((size_t)EXP_ROWS*D_)                            // 38400000
#define OFF_MAP    (OFF_SUB + 2*EXP_STRIDE)                         // 79359937

// top-k streaming buffer
#define CAP 128
#define TRIG 112   // prune when count exceeds this (<=16 inserts per tile per row)

typedef float v2f __attribute__((ext_vector_type(2)));
typedef float v8f __attribute__((ext_vector_type(8)));

__device__ __forceinline__ v8f wmma4(v2f a, v2f b, v8f c) {
  // V_WMMA_F32_16X16X4_F32 : D = A(16x4) * B(4x16) + C(16x16)
  return __builtin_amdgcn_wmma_f32_16x16x4_f32(false, a, false, b, (short)0, c,
                                               false, false);
}

// ---------------- item inverse norms ----------------
__global__ __launch_bounds__(256) void k_item_inv(const float* __restrict__ emb,
                                                  float* __restrict__ item_inv) {
  int wave = threadIdx.x >> 5, lane = threadIdx.x & 31;
  int row = blockIdx.x * 8 + wave;
  if (row >= NITEMS1) return;
  const float* p = emb + (size_t)(row + 1) * D_;
  float s = 0.f;
  for (int k = lane; k < D_; k += 32) { float v = p[k]; s = fmaf(v, v, s); }
  for (int off = 16; off; off >>= 1) s += __shfl_xor(s, off, 32);
  if (lane == 0) item_inv[row] = 1.0f / fmaxf(sqrtf(s), 1e-12f);
}

// ---------------- session embedding (one block per batch row) ----------------
__global__ __launch_bounds__(256) void k_sess(
    const float* __restrict__ emb, const float* __restrict__ posw,
    const float* __restrict__ w1, const float* __restrict__ w2,
    const float* __restrict__ glu1w, const float* __restrict__ glu1b,
    const float* __restrict__ glu2w, const int* __restrict__ inputs,
    const int* __restrict__ masks, float* __restrict__ sess_i) {
  __shared__ float seq[L_ * D_];    // gathered table[inputs[b]]
  __shared__ float nh1[L_ * D_];    // tanh(cat @ w1)
  __shared__ float hs[D_];          // masked mean, later reused as sess
  __shared__ float hg[D_];          // hs @ glu2_w
  __shared__ float beta[L_];
  __shared__ int   mk[L_];
  __shared__ float red[256];
  int b = blockIdx.x, t = threadIdx.x;

  if (t < L_) { mk[t] = masks[b * L_ + t]; beta[t] = 0.f; }
  for (int i = t; i < L_ * D_; i += 256) {
    int l = i / D_, d = i % D_;
    int idx = inputs[b * L_ + l];
    seq[i] = (idx == 0) ? 0.f : emb[(size_t)idx * D_ + d];  // table row0 is zero
  }
  __syncthreads();

  int cnt = 0;
  for (int l = 0; l < L_; ++l) cnt += mk[l];
  float inv = 1.0f / (float)cnt;
  for (int d = t; d < D_; d += 256) {
    float s = 0.f;
    for (int l = 0; l < L_; ++l) if (mk[l]) s += seq[l * D_ + d];
    hs[d] = s * inv;
  }
  __syncthreads();
  for (int d = t; d < D_; d += 256) {
    float s = 0.f;
    for (int k = 0; k < D_; ++k) s = fmaf(hs[k], glu2w[k * D_ + d], s);
    hg[d] = s;
  }
  // nh1 = tanh([pos, seq] @ w1)
  for (int i = t; i < L_ * D_; i += 256) {
    int l = i / D_, d = i % D_;
    float s = 0.f;
    for (int k = 0; k < D_; ++k) s = fmaf(posw[l * D_ + k], w1[k * D_ + d], s);
    for (int k = 0; k < D_; ++k) s = fmaf(seq[l * D_ + k], w1[(D_ + k) * D_ + d], s);
    nh1[i] = tanhf(s);
  }
  __syncthreads();
  // nh2 = sigmoid(nh1@glu1_w + glu1_b + hg);  beta[l] = sum_d nh2*w2[d]
  for (int i = t; i < L_ * D_; i += 256) {
    int l = i / D_, d = i % D_;
    float s = glu1b[d] + hg[d];
    for (int k = 0; k < D_; ++k) s = fmaf(nh1[l * D_ + k], glu1w[k * D_ + d], s);
    float sg = 1.0f / (1.0f + expf(-s));
    atomicAdd(&beta[l], sg * w2[d]);
  }
  __syncthreads();
  // sess[d] = sum_l beta[l]*mask*seq ; then 10*l2norm
  float np = 0.f;
  for (int d = t; d < D_; d += 256) {
    float s = 0.f;
    for (int l = 0; l < L_; ++l) if (mk[l]) s = fmaf(beta[l], seq[l * D_ + d], s);
    hs[d] = s;              // reuse hs as sess
    np = fmaf(s, s, np);
  }
  red[t] = np;
  __syncthreads();
  for (int s = 128; s; s >>= 1) { if (t < s) red[t] += red[t + s]; __syncthreads(); }
  float scale = 10.0f / fmaxf(sqrtf(red[0]), 1e-12f);
  for (int d = t; d < D_; d += 256) sess_i[b * D_ + d] = hs[d] * scale;
}

// ---------------- scores_item: sess_i @ l2norm(items)^T  (WMMA f32) ----------
__global__ __launch_bounds__(128) void k_scores(const float* __restrict__ sess_i,
                                                const float* __restrict__ emb,
                                                const float* __restrict__ item_inv,
                                                float* __restrict__ out) {
  int t = threadIdx.x, lane = t & 31, wave = t >> 5;
  int khalf = lane >> 4;
  int m0 = wave * 16;                       // 4 waves cover rows 0..63
  int col = blockIdx.x * 16 + (lane & 15);  // item column
  bool valid = col < NITEMS1;
  int colc = valid ? col : (NITEMS1 - 1);   // clamp: no divergence in K loop
  const float* brow = emb + (size_t)(colc + 1) * D_;
  const float* arow = sess_i + (m0 + (lane & 15)) * D_;
  v8f c = {};
  for (int k0 = 0; k0 < D_; k0 += 4) {
    int ka = k0 + 2 * khalf;
    v2f a, b;
    a.x = arow[ka];  a.y = arow[ka + 1];
    b.x = brow[ka];  b.y = brow[ka + 1];
    c = wmma4(a, b, c);
  }
  if (valid) {
    float scl = item_inv[col];
    for (int i = 0; i < 8; ++i) {
      int M = m0 + i + 8 * khalf;
      out[OFF_SCORES + (size_t)M * NITEMS1 + col] = c[i] * scl;
    }
  }
}

// ---------------- cross-entropy loss over 39999 classes ----------------
__global__ __launch_bounds__(256) void k_loss_row(const float* __restrict__ out,
                                                  const int* __restrict__ targets,
                                                  float* __restrict__ row_loss) {
  __shared__ float red[256];
  int b = blockIdx.x, t = threadIdx.x;
  const float* row = out + OFF_SCORES + (size_t)b * NITEMS1;
  float mx = -FLT_MAX;
  for (int n = t; n < NITEMS1; n += 256) mx = fmaxf(mx, row[n]);
  red[t] = mx; __syncthreads();
  for (int s = 128; s; s >>= 1) { if (t < s) red[t] = fmaxf(red[t], red[t + s]); __syncthreads(); }
  mx = red[0]; __syncthreads();
  float sum = 0.f;
  for (int n = t; n < NITEMS1; n += 256) sum += expf(row[n] - mx);
  red[t] = sum; __syncthreads();
  for (int s = 128; s; s >>= 1) { if (t < s) red[t] += red[t + s]; __syncthreads(); }
  if (t == 0) {
    int lbl = targets[b] - 1;
    row_loss[b] = (mx + logf(red[0])) - row[lbl];   // -logp[label]
  }
}

__global__ __launch_bounds__(64) void k_loss_final(const float* __restrict__ row_loss,
                                                   float* __restrict__ out) {
  __shared__ float red[64];
  int t = threadIdx.x;
  red[t] = row_loss[t]; __syncthreads();
  for (int s = 32; s; s >>= 1) { if (t < s) red[t] += red[t + s]; __syncthreads(); }
  if (t == 0) out[OFF_LOSS] = red[0] / (float)B_;
}

// ---------------- fused sim-GEMM + exact streaming top-100 -------------------
// One wave32 per 16-row tile. Streams all 39999 items (L2-resident) through
// V_WMMA_F32_16X16X4_F32, maintains per-row top-100 via threshold + buffer.
__global__ __launch_bounds__(32) void k_sim_topk(const float* __restrict__ emb,
                                                 const int* __restrict__ viewed,
                                                 const int* __restrict__ bought,
                                                 int* __restrict__ sel,
                                                 const int* __restrict__ slices) {
  __shared__ float At[16 * 304];       // A tile, padded pitch
  __shared__ float tv[16 * CAP];       // candidate values
  __shared__ int   ti[16 * CAP];       // candidate item indices
  __shared__ int   tc[16];             // counts
  __shared__ float thr[16];            // running 100th-largest threshold

  int lane = threadIdx.x;
  int tile = blockIdx.x;               // 0..799
  int source = (tile >= TILES_PER_SRC) ? 1 : 0;
  int rt = (tile % TILES_PER_SRC) * 16;          // base row within source
  const int* src = source ? bought : viewed;     // flat (B,L,S)

  for (int i = lane; i < 16 * D_; i += 32) {
    int r = i / D_, d = i % D_;
    int idx = src[rt + r];             // raw embedding_weight gather (incl row 0)
    At[r * 304 + d] = emb[(size_t)idx * D_ + d];
  }
  if (lane < 16) { tc[lane] = 0; thr[lane] = -FLT_MAX; }
  __syncthreads();

  int khalf = lane >> 4;
  int colb = lane & 15;
  int apitch = (lane & 15) * 304;
  for (int n0 = 0; n0 < NITEMS1; n0 += 16) {
    int col = n0 + colb;
    bool valid = col < NITEMS1;
    int colc = valid ? col : (NITEMS1 - 1);      // clamp: branch-free K loop
    const float* brow = emb + (size_t)(colc + 1) * D_;
    // prefetch next tile's item row while this tile computes
    if (n0 + 16 < NITEMS1)
      __builtin_prefetch(emb + (size_t)(colc + 17) * D_, 0, 0);
    v8f c = {};
    for (int k0 = 0; k0 < D_; k0 += 4) {
      int ka = k0 + 2 * khalf;
      v2f a, b;
      a.x = At[apitch + ka];  a.y = At[apitch + ka + 1];
      b.x = brow[ka];         b.y = brow[ka + 1];
      c = wmma4(a, b, c);
    }
    // insert candidates that beat the running threshold
    for (int i = 0; i < 8; ++i) {
      int r = i + 8 * khalf;
      float v = c[i];
      if (valid && v > thr[r]) {
        int slot = atomicAdd(&tc[r], 1);   // bounded: pre-tile count <= TRIG
        tv[r * CAP + slot] = v;
        ti[r * CAP + slot] = col;
      }
    }
    __syncthreads();
    // prune any row close to capacity back down to exact top-100
    for (int r = 0; r < 16; ++r) {
      if (tc[r] > TRIG) {
        float* tvr = tv + r * CAP;
        int*   tir = ti + r * CAP;
        int cnt = tc[r];
        while (cnt > COARSE_) {
          float bv = FLT_MAX; int bs = -1;
          for (int j = lane; j < cnt; j += 32) { float v = tvr[j]; if (v < bv) { bv = v; bs = j; } }
          for (int off = 16; off; off >>= 1) {
            float ov = __shfl_xor(bv, off, 32); int os = __shfl_xor(bs, off, 32);
            if (ov < bv) { bv = ov; bs = os; }
          }
          __syncthreads();
          if (lane == 0) { tvr[bs] = tvr[cnt - 1]; tir[bs] = tir[cnt - 1]; }
          __syncthreads();
          --cnt;
        }
        float bv = FLT_MAX;
        for (int j = lane; j < cnt; j += 32) bv = fminf(bv, tvr[j]);
        for (int off = 16; off; off >>= 1) bv = fminf(bv, __shfl_xor(bv, off, 32));
        __syncthreads();
        if (lane == 0) { tc[r] = cnt; thr[r] = bv; }
        __syncthreads();
      }
    }
    __syncthreads();
  }

  // final: rank the top-100 descending (ties -> smaller index, like top_k)
  int* rank = (int*)At;                 // reuse A-tile LDS: 16 rows x 128 ints
  for (int r = 0; r < 16; ++r) {
    float* tvr = tv + r * CAP;
    int*   tir = ti + r * CAP;
    int cnt = tc[r];                    // in [100, CAP]
    for (int rk = 0; rk < COARSE_; ++rk) {
      float bv = -FLT_MAX; int bi = 0x7fffffff; int bs = -1;
      for (int j = lane; j < cnt; j += 32) {
        float v = tvr[j]; int id = tir[j];
        if (v > bv || (v == bv && id < bi)) { bv = v; bi = id; bs = j; }
      }
      for (int off = 16; off; off >>= 1) {
        float ov = __shfl_xor(bv, off, 32);
        int   oi = __shfl_xor(bi, off, 32);
        int   os = __shfl_xor(bs, off, 32);
        if (ov > bv || (ov == bv && oi < bi)) { bv = ov; bi = oi; bs = os; }
      }
      __syncthreads();
      if (lane == 0) { rank[r * 128 + rk] = bi; tvr[bs] = tvr[cnt - 1]; tir[bs] = tir[cnt - 1]; }
      __syncthreads();
      --cnt;
    }
  }
  __syncthreads();
  for (int i = lane; i < 16 * EXP_; i += 32) {
    int r = i / EXP_, j = i % EXP_;
    int gr = source * ROWS_PER_SRC + rt + r;
    sel[(size_t)gr * EXP_ + j] = rank[r * 128 + slices[j]];
  }
}

// ---------------- gathered-A GEMM: all_item[sel] @ map_dense -----------------
__global__ __launch_bounds__(128) void k_expand(const float* __restrict__ emb,
                                                const float* __restrict__ mapd,
                                                const int* __restrict__ sel,
                                                float* __restrict__ out) {
  int t = threadIdx.x, lane = t & 31, wave = t >> 5;
  long task = (long)blockIdx.x * 4 + wave;      // < 2*8000*19
  int source = (int)(task / ((long)M_TILES_EXP * N_TILES_EXP));
  int tt = (int)(task % ((long)M_TILES_EXP * N_TILES_EXP));
  int m0 = (tt / N_TILES_EXP) * 16;
  int n0 = (tt % N_TILES_EXP) * 16;
  int khalf = lane >> 4;
  int mrow = m0 + (lane & 15);                  // 0..127999, = p*100 + s*20 + j
  int r = mrow / EXP_, j = mrow % EXP_;         // r = (b*L+l)*S+s
  int item = sel[((size_t)source * ROWS_PER_SRC + r) * EXP_ + j];
  const float* arow = emb + (size_t)(item + 1) * D_;
  int ncol = n0 + (lane & 15);
  bool nvalid = ncol < D_;
  int ncc = nvalid ? ncol : (D_ - 1);           // clamp: branch-free K loop
  v8f c = {};
  for (int k0 = 0; k0 < D_; k0 += 4) {
    int ka = k0 + 2 * khalf;
    v2f a, b;
    a.x = arow[ka];               a.y = arow[ka + 1];
    b.x = mapd[ka * D_ + ncc];    b.y = mapd[(ka + 1) * D_ + ncc];
    c = wmma4(a, b, c);
  }
  if (nvalid) {
    size_t base = OFF_SUB + (size_t)source * EXP_STRIDE;
    // streamed once, never re-read -> non-temporal to keep item table in L2
    for (int i = 0; i < 8; ++i) {
      int M = m0 + i + 8 * khalf;
      __builtin_nontemporal_store(c[i], &out[base + (size_t)M * D_ + ncol]);
    }
  }
}

// ---------------- sess_mapped = sess_i @ map_dense ----------------
__global__ __launch_bounds__(320) void k_sess_map(const float* __restrict__ sess_i,
                                                  const float* __restrict__ mapd,
                                                  float* __restrict__ out) {
  int b = blockIdx.x, d = threadIdx.x;
  if (d >= D_) return;
  float s = 0.f;
  for (int k = 0; k < D_; ++k) s = fmaf(sess_i[b * D_ + k], mapd[k * D_ + d], s);
  out[OFF_MAP + (size_t)b * D_ + d] = s;
}

// ---------------- host wiring ----------------
extern "C" void kernel_launch(void* const* d_in, const int* in_sizes, int n_in,
                              void* d_out, int out_size, void* d_ws, size_t ws_size,
                              hipStream_t stream) {
  const float* emb    = (const float*)d_in[0];
  const float* posw   = (const float*)d_in[1];
  const float* w1     = (const float*)d_in[2];
  const float* w2     = (const float*)d_in[3];
  const float* glu1w  = (const float*)d_in[4];
  const float* glu1b  = (const float*)d_in[5];
  const float* glu2w  = (const float*)d_in[6];
  const float* mapd   = (const float*)d_in[7];
  const int*   inputs = (const int*)d_in[8];
  const int*   targets= (const int*)d_in[9];
  const int*   masks  = (const int*)d_in[10];
  const int*   viewed = (const int*)d_in[11];
  const int*   bought = (const int*)d_in[12];
  const int*   slices = (const int*)d_in[13];
  float* out = (float*)d_out;
  float* ws  = (float*)d_ws;

  // workspace layout (floats): [0,19200) sess_i | [19200,59200) item_inv |
  // [59200,59264) row_loss | then sel as int[2*6400*20]
  float* sess_i   = ws;
  float* item_inv = ws + 19200;
  float* row_loss = ws + 59200;
  int*   sel      = (int*)(ws + 59264);

  k_item_inv<<<(NITEMS1 + 7) / 8, 256, 0, stream>>>(emb, item_inv);
  k_sess<<<B_, 256, 0, stream>>>(emb, posw, w1, w2, glu1w, glu1b, glu2w,
                                 inputs, masks, sess_i);
  k_scores<<<(NITEMS1 + 15) / 16, 128, 0, stream>>>(sess_i, emb, item_inv, out);
  k_loss_row<<<B_, 256, 0, stream>>>(out, targets, row_loss);
  k_loss_final<<<1, 64, 0, stream>>>(row_loss, out);
  k_sim_topk<<<2 * TILES_PER_SRC, 32, 0, stream>>>(emb, viewed, bought, sel, slices);
  k_expand<<<(2 * M_TILES_EXP * N_TILES_EXP) / 4, 128, 0, stream>>>(emb, mapd, sel, out);
  k_sess_map<<<B_, 320, 0, stream>>>(sess_i, mapd, out);
}